// La_aggregator_85899346058
// MI455X (gfx1250) — compile-verified
//
#include <hip/hip_runtime.h>
#include <stdint.h>

typedef float        __attribute__((ext_vector_type(2))) v2f;
typedef float        __attribute__((ext_vector_type(8))) v8f;
typedef unsigned int __attribute__((ext_vector_type(4))) v4u;
typedef int          __attribute__((ext_vector_type(8))) v8i;
typedef int          __attribute__((ext_vector_type(4))) v4i;

#define BATCH   8
#define AGENTS  256
#define DIM     128
#define I_PER_WG 8
#define WGS_PER_B (AGENTS / I_PER_WG)   // 32

// ---------------------------------------------------------------------------
// Kernel 1: Tj = nf @ W[:, :D]^T ; Ti = nf @ W[:, D:]^T + bias
// One wave32 computes one 16x16 fp32 tile via v_wmma_f32_16x16x4_f32.
// f32 A-matrix 16x4 layout (ISA 7.12.2): lanes 0-15 hold K={0,1} in the two
// VGPRs, lanes 16-31 hold K={2,3}. B 4x16 mirrors it (rows striped on lanes).
// ---------------------------------------------------------------------------
__global__ __launch_bounds__(256) void la_lin_wmma_kernel(
    const float* __restrict__ nf,    // [B, A, D]
    const float* __restrict__ W,     // [D, 2D] row-major, row length 2D
    const float* __restrict__ bias,  // [D]
    float* __restrict__ Tj,          // [B, A, D]
    float* __restrict__ Ti)          // [B, A, D] (bias folded in)
{
    const int tid  = threadIdx.x;
    const int lane = tid & 31;
    const int lo   = lane & 15;
    const int hi   = lane >> 4;                 // 0: K base +0, 1: K base +2
    const int gt   = blockIdx.x * 8 + (tid >> 5);   // tile id 0..2047
    const int b    = gt >> 8;
    const int r    = gt & 255;
    const int isTi = r >> 7;                    // 0 -> Tj tile, 1 -> Ti tile
    const int t    = r & 127;                   // 16 mtiles x 8 ntiles
    const int m0   = (t >> 3) << 4;
    const int e0   = (t & 7) << 4;

    // A[m][k] = nf[b, m, k]; this lane supplies rows m0+lo.
    const float* Arow = nf + (size_t)(b * AGENTS + m0 + lo) * DIM;
    // B[k][n] = W[e0+n][ (isTi?D:0) + k ]; this lane supplies column n = lo.
    const float* Brow = W + (size_t)(e0 + lo) * (2 * DIM) + (isTi ? DIM : 0);

    v8f c = {};
    #pragma unroll
    for (int k0 = 0; k0 < DIM; k0 += 4) {
        const int k = k0 + (hi << 1);
        v2f a, bb;
        a.x  = Arow[k];     a.y  = Arow[k + 1];
        bb.x = Brow[k];     bb.y = Brow[k + 1];
        c = __builtin_amdgcn_wmma_f32_16x16x4_f32(
                /*neg_a=*/false, a, /*neg_b=*/false, bb,
                /*c_mod=*/(short)0, c, /*reuse_a=*/false, /*reuse_b=*/false);
    }

    float* dst = isTi ? Ti : Tj;
    const float bv = isTi ? bias[e0 + lo] : 0.0f;
    // C/D layout: vgpr v holds row m0+v (lanes 0-15) / m0+8+v (lanes 16-31),
    // column e0+lo in both halves.
    #pragma unroll
    for (int v = 0; v < 8; ++v) {
        const int row = m0 + (hi << 3) + v;
        dst[(size_t)(b * AGENTS + row) * DIM + e0 + lo] = c[v] + bv;
    }
}

// ---------------------------------------------------------------------------
// TDM: build a 2D D# (ISA §8.3/§8.4) and issue TENSOR_LOAD_TO_LDS.
// 4B elements; tile = tile_d0 x tile_d1, row stride stride0 (elements).
// 6-arg builtin on this toolchain: (v4u g0, v8i g1, v4i g2, v4i g3, v8i g4, i32 cpol)
// ---------------------------------------------------------------------------
#if __has_builtin(__builtin_amdgcn_tensor_load_to_lds) && \
    __has_builtin(__builtin_amdgcn_s_wait_tensorcnt)
#define USE_TDM 1
__device__ __forceinline__ void tdm_load_2d(uint32_t lds_off, const void* gptr,
                                            uint32_t tile_d0, uint32_t tile_d1,
                                            uint32_t stride0)
{
    const uint64_t g = (uint64_t)gptr;
    const uint32_t td0 = stride0;     // tensor_dim0 (elements)
    const uint32_t td1 = tile_d1;     // tensor_dim1

    v4u g0;
    g0[0] = 1u;                                        // count=1, user mode
    g0[1] = lds_off;                                   // lds_addr (bytes)
    g0[2] = (uint32_t)g;                               // global_addr[31:0]
    g0[3] = (uint32_t)((g >> 32) & 0x01FFFFFFu)        // global_addr[56:32]
          | (2u << 30);                                // type = 2 ("image")

    v8i g1;
    g1[0] = 0x20000;                                   // data_size=2 (4B), mask=0
    g1[1] = (int)((td0 & 0xFFFFu) << 16);              // tensor_dim0[15:0]
    g1[2] = (int)((td0 >> 16) | ((td1 & 0xFFFFu) << 16)); // td0 hi | td1 lo
    g1[3] = (int)((td1 >> 16) | ((tile_d0 & 0xFFFFu) << 16)); // td1 hi | tile_dim0
    g1[4] = (int)(tile_d1 & 0xFFFFu);                  // tile_dim1, tile_dim2=0
    g1[5] = (int)stride0;                              // tensor_dim0_stride[31:0]
    g1[6] = 0;                                         // stride0 hi, stride1 lo
    g1[7] = 0;                                         // stride1 hi
    v4i g2 = {0, 0, 0, 0};
    v4i g3 = {0, 0, 0, 0};
    v8i g4 = {0, 0, 0, 0, 0, 0, 0, 0};
    __builtin_amdgcn_tensor_load_to_lds(g0, g1, g2, g3, g4, 0);
}
#else
#define USE_TDM 0
#endif

// ---------------------------------------------------------------------------
// Kernel 2: fused aggregation.
// out[b,i,e] = sum_j sigmoid((Ti[b,i,e]+Tj[b,j,e])*(adj[b,i,j]+[i==j])) * nf[b,j,e]
// One workgroup = one batch b + 8 rows i (one wave per row).  Each lane owns
// 4 consecutive e, so the inner loop is 2x ds_load_b128 + 1 broadcast b32
// feeding 4 sigmoid+FMA chains.  Tj[b], nf[b], adj rows staged in LDS by TDM.
// ---------------------------------------------------------------------------
__global__ __launch_bounds__(256) void la_agg_kernel(
    const float* __restrict__ nf,    // [B, A, D]
    const float* __restrict__ adj,   // [B, A, A]
    const float* __restrict__ Tj,    // [B, A, D]
    const float* __restrict__ Ti,    // [B, A, D] (bias included)
    float* __restrict__ out)         // [B, A, D]
{
    extern __shared__ float smem[];
    float* sTj  = smem;                        // AGENTS*DIM  (128 KB)
    float* sNf  = smem +     AGENTS * DIM;     // AGENTS*DIM  (128 KB)
    float* sAdj = smem + 2 * AGENTS * DIM;     // I_PER_WG*AGENTS (8 KB)

    const int tid = threadIdx.x;
    const int b   = blockIdx.x / WGS_PER_B;
    const int i0  = (blockIdx.x % WGS_PER_B) * I_PER_WG;

    const float* gTj  = Tj  + (size_t)b * AGENTS * DIM;
    const float* gNf  = nf  + (size_t)b * AGENTS * DIM;
    const float* gAdj = adj + (size_t)b * AGENTS * AGENTS + (size_t)i0 * AGENTS;

#if USE_TDM
    if (tid < 32) {   // wave 0 drives the three DMAs
        tdm_load_2d((uint32_t)(uintptr_t)sTj,  gTj,  DIM,    AGENTS,   DIM);
        tdm_load_2d((uint32_t)(uintptr_t)sNf,  gNf,  DIM,    AGENTS,   DIM);
        tdm_load_2d((uint32_t)(uintptr_t)sAdj, gAdj, AGENTS, I_PER_WG, AGENTS);
        __builtin_amdgcn_s_wait_tensorcnt(0);
    }
#else
    for (int idx = tid; idx < AGENTS * DIM; idx += 256) {
        sTj[idx] = gTj[idx];
        sNf[idx] = gNf[idx];
    }
    for (int idx = tid; idx < I_PER_WG * AGENTS; idx += 256) sAdj[idx] = gAdj[idx];
#endif
    __syncthreads();

    const int wave = tid >> 5;                 // 0..7 -> row i0+wave (uniform/wave)
    const int lane = tid & 31;                 // lane owns e = 4*lane .. 4*lane+3
    const int i    = i0 + wave;

    const float4* sTj4 = (const float4*)sTj;   // [j][lane] of float4
    const float4* sNf4 = (const float4*)sNf;

    const float4 ti4 = *(const float4*)(Ti + (size_t)(b * AGENTS + i) * DIM + 4 * lane);

    float4 acc = make_float4(0.f, 0.f, 0.f, 0.f);
    #pragma unroll 4
    for (int j = 0; j < AGENTS; ++j) {
        const float  a   = sAdj[wave * AGENTS + j] + ((j == i) ? 1.0f : 0.0f);
        const float4 tj4 = sTj4[j * 32 + lane];
        const float4 nf4 = sNf4[j * 32 + lane];
        const float sx = __builtin_amdgcn_rcpf(1.0f + __expf(-(ti4.x + tj4.x) * a));
        const float sy = __builtin_amdgcn_rcpf(1.0f + __expf(-(ti4.y + tj4.y) * a));
        const float sz = __builtin_amdgcn_rcpf(1.0f + __expf(-(ti4.z + tj4.z) * a));
        const float sw = __builtin_amdgcn_rcpf(1.0f + __expf(-(ti4.w + tj4.w) * a));
        acc.x += sx * nf4.x;
        acc.y += sy * nf4.y;
        acc.z += sz * nf4.z;
        acc.w += sw * nf4.w;
    }
    *(float4*)(out + (size_t)(b * AGENTS + i) * DIM + 4 * lane) = acc;
}

// ---------------------------------------------------------------------------
extern "C" void kernel_launch(void* const* d_in, const int* in_sizes, int n_in,
                              void* d_out, int out_size, void* d_ws, size_t ws_size,
                              hipStream_t stream)
{
    const float* nf   = (const float*)d_in[0];  // [8,256,128]
    /* d_in[1] = nodes (int64) — unused by forward */
    const float* adj  = (const float*)d_in[2];  // [8,256,256]
    const float* W    = (const float*)d_in[3];  // [128,256]
    const float* bias = (const float*)d_in[4];  // [128]
    float* out = (float*)d_out;

    float* Tj = (float*)d_ws;                          // 1 MB
    float* Ti = Tj + (size_t)BATCH * AGENTS * DIM;     // 1 MB

    // 2048 tiles / 8 waves per block = 256 blocks
    la_lin_wmma_kernel<<<256, 256, 0, stream>>>(nf, W, bias, Tj, Ti);

    const size_t shmem = (size_t)(2 * AGENTS * DIM + I_PER_WG * AGENTS) * sizeof(float);
    la_agg_kernel<<<BATCH * WGS_PER_B, 256, shmem, stream>>>(nf, adj, Tj, Ti, out);
}